// GraphAttentionLayer_83459804496006
// MI455X (gfx1250) — compile-verified
//
#include <hip/hip_runtime.h>
#include <hip/hip_bf16.h>
#include <cstddef>
#include <cstdint>

// ---------------------------------------------------------------------------
// GAT layer for MI455X (gfx1250, wave32, WMMA + async global->LDS)
//   B=4, N=1024, F_IN=128, HEADS=8, HID=32, OUT=256
// ---------------------------------------------------------------------------

#define B_DIM 4
#define N_DIM 1024
#define F_IN 128
#define F_OUT 256
#define HEADS 8
#define HID 32
#define ROWS (B_DIM * N_DIM)          // 4096
#define NEG_SLOPE 0.2f

typedef float v2f __attribute__((ext_vector_type(2)));
typedef float v8f __attribute__((ext_vector_type(8)));

__device__ __forceinline__ float lrelu(float x) {
    return x >= 0.0f ? x : NEG_SLOPE * x;
}

// ---- CDNA5 async global->LDS copy (GLOBAL_LOAD_ASYNC_TO_LDS_B128) ---------
// Per-lane gather: each lane copies 16 bytes from global address gaddr to LDS offset lds_off, tracked by
// ASYNCcnt. lds_off is the low-32-bit LDS byte offset (generic shared
// pointers carry the LDS offset in their low 32 bits on gfx1250).
__device__ __forceinline__ void async_copy_b128(unsigned lds_off, const void* gaddr) {
    asm volatile("global_load_async_to_lds_b128 %0, %1, off"
                 :: "v"(lds_off), "v"(gaddr)
                 : "memory");
}
__device__ __forceinline__ void wait_asynccnt0() {
    asm volatile("s_wait_asynccnt 0x0" ::: "memory");
}
__device__ __forceinline__ unsigned lds_off_of(const void* shared_ptr) {
    return (unsigned)(uintptr_t)shared_ptr;
}

// ---------------------------------------------------------------------------
// Kernel 1: g = h @ W   (4096x128 * 128x256 -> 4096x256), fp32 WMMA 16x16x4.
// One wave per 16x16 output tile. 8 waves (256 threads) per block.
// ---------------------------------------------------------------------------
__global__ void k_gemm_g(const float* __restrict__ h,
                         const float* __restrict__ W,
                         float* __restrict__ g) {
    const int lane  = threadIdx.x & 31;
    const int wid   = blockIdx.x * 8 + (threadIdx.x >> 5);   // 0..4095
    const int mtile = wid >> 4;                              // 0..255
    const int ntile = wid & 15;                              // 0..15

    const int il    = lane & 15;
    const int khalf = (lane >> 4) << 1;                      // 0 or 2
    const float* hrow = h + (size_t)((mtile << 4) + il) * F_IN;
    const int ncol  = (ntile << 4) + il;

    v8f c = {};
    #pragma unroll 4
    for (int k = 0; k < F_IN; k += 4) {
        v2f a, b;
        a.x = hrow[k + khalf];
        a.y = hrow[k + khalf + 1];
        b.x = W[(size_t)(k + khalf) * F_OUT + ncol];
        b.y = W[(size_t)(k + khalf + 1) * F_OUT + ncol];
        c = __builtin_amdgcn_wmma_f32_16x16x4_f32(false, a, false, b,
                                                  (short)0, c, false, false);
    }

    const int mbase = (mtile << 4) + ((lane >> 4) << 3);
    #pragma unroll
    for (int r = 0; r < 8; ++r)
        g[(size_t)(mbase + r) * F_OUT + ncol] = c[r];
}

// ---------------------------------------------------------------------------
// Kernel 2: el[row,h] = dot(g[row,h,:], a_l) ; er likewise with a_r.
// ---------------------------------------------------------------------------
__global__ void k_el_er(const float* __restrict__ g,
                        const float* __restrict__ a_vec,
                        float* __restrict__ el,
                        float* __restrict__ er) {
    const int t = blockIdx.x * blockDim.x + threadIdx.x;     // 0..32767
    if (t >= ROWS * HEADS) return;
    const int row = t >> 3;
    const int hh  = t & 7;
    const float* gp = g + (size_t)row * F_OUT + hh * HID;
    float sl = 0.0f, sr = 0.0f;
    #pragma unroll 8
    for (int f = 0; f < HID; ++f) {
        const float v = gp[f];
        sl += v * a_vec[f];
        sr += v * a_vec[HID + f];
    }
    el[t] = sl;
    er[t] = sr;
}

// ---------------------------------------------------------------------------
// Kernel 3: per (b,i,h) masked-softmax stats over j.
// adj row (4KB) + er slab (32KB) staged via async global->LDS.
// ---------------------------------------------------------------------------
__global__ void k_softmax_stats(const int* __restrict__ adj,
                                const float* __restrict__ el,
                                const float* __restrict__ er,
                                float* __restrict__ mrow,
                                float* __restrict__ srow) {
    __shared__ int   adjrow[N_DIM];
    __shared__ float ersh[N_DIM * HEADS];

    const int b = blockIdx.x >> 10;
    const int i = blockIdx.x & (N_DIM - 1);

    // async stage: adj row = 256 x 16B chunks, er slab = 2048 x 16B chunks
    const int*   arow = adj + ((size_t)(b * N_DIM + i)) * N_DIM;
    const float* erb  = er + (size_t)b * N_DIM * HEADS;
    {
        const int t = threadIdx.x;                      // one chunk each
        async_copy_b128(lds_off_of(&adjrow[t * 4]), arow + t * 4);
        #pragma unroll
        for (int k = 0; k < 8; ++k) {
            const int cidx = t + k * 256;               // 0..2047
            async_copy_b128(lds_off_of(&ersh[cidx * 4]), erb + cidx * 4);
        }
    }
    wait_asynccnt0();
    __syncthreads();

    const int h    = threadIdx.x >> 5;
    const int lane = threadIdx.x & 31;
    const float eli = el[(size_t)(b * N_DIM + i) * HEADS + h];

    float m = -3.0e38f, s = 0.0f;
    for (int j = lane; j < N_DIM; j += 32) {
        if (adjrow[j]) {
            float e = lrelu(eli + ersh[j * HEADS + h]);
            if (e > m) {
                s = s * __expf(m - e) + 1.0f;
                m = e;
            } else {
                s += __expf(e - m);
            }
        }
    }
    // wave32 tree combine of (m, s)
    #pragma unroll
    for (int off = 16; off > 0; off >>= 1) {
        const float mo = __shfl_xor(m, off, 32);
        const float so = __shfl_xor(s, off, 32);
        const float M  = fmaxf(m, mo);
        s = s * __expf(m - M) + so * __expf(mo - M);
        m = M;
    }
    if (lane == 0) {
        const size_t idx = (size_t)(b * N_DIM + i) * HEADS + h;
        mrow[idx] = m;
        srow[idx] = s;
    }
}

// ---------------------------------------------------------------------------
// Kernel 4: out[b,i,h,:] = lrelu( sum_j a[b,i,j,h] * g[b,j,h,:] )
// One block per (b, i-tile of 16); 8 waves = 8 heads. adj tile staged as
// padded bytes (int4 load -> packed u32 store); er slab staged via async
// global->LDS. Attention weights rebuilt in registers in WMMA A-fragment
// layout; two 16x16 f32 accumulators per head; K over j in steps of 4.
// ---------------------------------------------------------------------------
#define ADJ_PITCH 1032   // bytes per adj row in LDS (4-aligned, kills conflicts)

__global__ void k_attn(const int* __restrict__ adj,
                       const float* __restrict__ g,
                       const float* __restrict__ el,
                       const float* __restrict__ er,
                       const float* __restrict__ mrow,
                       const float* __restrict__ srow,
                       float* __restrict__ out) {
    __shared__ unsigned char adjt[16 * ADJ_PITCH];
    __shared__ float ersh[N_DIM * HEADS];

    const int b     = blockIdx.x >> 6;          // 0..3
    const int itile = blockIdx.x & 63;          // 0..63

    // async stage er slab (2048 x 16B chunks)
    const float* erb = er + (size_t)b * N_DIM * HEADS;
    #pragma unroll
    for (int k = 0; k < 8; ++k) {
        const int cidx = threadIdx.x + k * 256;
        async_copy_b128(lds_off_of(&ersh[cidx * 4]), erb + cidx * 4);
    }

    // adj tile -> bytes in LDS (int4 loads, packed u32 stores)
    const int* abase = adj + ((size_t)(b * N_DIM + itile * 16)) * N_DIM;
    for (int t = threadIdx.x; t < 16 * (N_DIM / 4); t += 256) {
        const int ilr = t >> 8;                 // row 0..15
        const int j4  = (t & 255) << 2;         // j = 0,4,...,1020
        const int4 v = ((const int4*)(abase + (size_t)ilr * N_DIM))[t & 255];
        const unsigned p = (unsigned)(v.x != 0)
                         | ((unsigned)(v.y != 0) << 8)
                         | ((unsigned)(v.z != 0) << 16)
                         | ((unsigned)(v.w != 0) << 24);
        *(unsigned*)&adjt[ilr * ADJ_PITCH + j4] = p;
    }
    wait_asynccnt0();
    __syncthreads();

    const int h     = threadIdx.x >> 5;
    const int lane  = threadIdx.x & 31;
    const int il    = lane & 15;
    const int khalf = (lane >> 4) << 1;         // 0 or 2

    const size_t rbase = (size_t)(b * N_DIM + itile * 16 + il) * HEADS + h;
    const float eli   = el[rbase];
    const float mi    = mrow[rbase];
    const float inv_s = 1.0f / srow[rbase];

    const float* gb = g + (size_t)b * N_DIM * F_OUT + h * HID;
    const unsigned char* arow = adjt + il * ADJ_PITCH;

    v8f c0 = {}, c1 = {};
    for (int j0 = 0; j0 < N_DIM; j0 += 4) {
        const int ja = j0 + khalf;

        v2f a;
        {
            const float e0 = lrelu(eli + ersh[ja * HEADS + h]);
            a.x = arow[ja] ? __expf(e0 - mi) * inv_s : 0.0f;
            const float e1 = lrelu(eli + ersh[(ja + 1) * HEADS + h]);
            a.y = arow[ja + 1] ? __expf(e1 - mi) * inv_s : 0.0f;
        }

        const float* g0 = gb + (size_t)ja * F_OUT;   // row j=ja of g (this head)
        v2f b0, b1;
        b0.x = g0[il];              // f = 0..15, K = ja
        b0.y = g0[F_OUT + il];      //              K = ja+1
        b1.x = g0[16 + il];         // f = 16..31
        b1.y = g0[F_OUT + 16 + il];

        c0 = __builtin_amdgcn_wmma_f32_16x16x4_f32(false, a, false, b0,
                                                   (short)0, c0, false, false);
        c1 = __builtin_amdgcn_wmma_f32_16x16x4_f32(false, a, false, b1,
                                                   (short)0, c1, false, false);
    }

    const int mhalf = (lane >> 4) << 3;
    #pragma unroll
    for (int r = 0; r < 8; ++r) {
        const int orow = itile * 16 + mhalf + r;
        const size_t obase = (size_t)(b * N_DIM + orow) * F_OUT + h * HID;
        out[obase + il]      = lrelu(c0[r]);
        out[obase + 16 + il] = lrelu(c1[r]);
    }
}

// ---------------------------------------------------------------------------
// Launcher. Inputs: h (f32), adj_mat (i32), W (f32), a_vec (f32).
// Workspace layout (floats): g[4096*256] | el[32768] | er[32768]
//                            | mrow[32768] | srow[32768]  (~4.7 MB)
// ---------------------------------------------------------------------------
extern "C" void kernel_launch(void* const* d_in, const int* in_sizes, int n_in,
                              void* d_out, int out_size, void* d_ws, size_t ws_size,
                              hipStream_t stream) {
    const float* h     = (const float*)d_in[0];
    const int*   adj   = (const int*)d_in[1];
    const float* W     = (const float*)d_in[2];
    const float* a_vec = (const float*)d_in[3];
    float* out = (float*)d_out;

    float* wsf  = (float*)d_ws;
    float* g    = wsf;
    float* el   = g + (size_t)ROWS * F_OUT;
    float* er   = el + ROWS * HEADS;
    float* mrow = er + ROWS * HEADS;
    float* srow = mrow + ROWS * HEADS;

    // 1) g = h @ W   (4096 tiles / 8 waves per block = 512 blocks)
    hipLaunchKernelGGL(k_gemm_g, dim3(512), dim3(256), 0, stream, h, W, g);
    // 2) el / er
    hipLaunchKernelGGL(k_el_er, dim3((ROWS * HEADS) / 256), dim3(256), 0, stream,
                       g, a_vec, el, er);
    // 3) softmax stats (one block per (b,i))
    hipLaunchKernelGGL(k_softmax_stats, dim3(B_DIM * N_DIM), dim3(256), 0, stream,
                       adj, el, er, mrow, srow);
    // 4) attention aggregation + output activation
    hipLaunchKernelGGL(k_attn, dim3(B_DIM * (N_DIM / 16)), dim3(256), 0, stream,
                       adj, g, el, er, mrow, srow, out);
}